// FocalLoss_72430328481122
// MI455X (gfx1250) — compile-verified
//
#include <hip/hip_runtime.h>
#include <hip/hip_bf16.h>
#include <math.h>

// ---------------------------------------------------------------------------
// FocalLoss (RetinaNet) for MI455X / gfx1250, wave32.
// One fused HBM pass: IoU assignment + focal BCE + smooth-L1, then
// WMMA-based (v_wmma_f32_16x16x4_f32) wave reduction -> per-image atomics ->
// tiny finalize kernel producing the 2 output scalars.
// Annotations staged into LDS via gfx1250 async global->LDS DMA (ASYNCcnt).
// ---------------------------------------------------------------------------

typedef float v2f __attribute__((ext_vector_type(2)));
typedef float v8f __attribute__((ext_vector_type(8)));

#define MAX_M 32   // LDS capacity for annotations (M=24 in reference)

// Exact 32-lane f32 sum using the f32-native WMMA shape 16x16x4.
// A-matrix layout (32-bit, 16x4): lane 0-15: VGPR0->K0, VGPR1->K1;
//                                 lane 16-31: VGPR0->K2, VGPR1->K3; M = lane%16.
// With A[m][0]=p[lane m], A[m][2]=p[lane m+16], A[m][1]=A[m][3]=0 and B=ones:
//   D[m][n] = p[m] + p[m+16]  (rowsum, replicated over n)
// D layout: VGPR r holds rowsum[r] in lanes 0-15 and rowsum[r+8] in lanes 16-31,
// so sum(d0..d7) per lane = half-sum; one shfl_xor(16) finishes the reduction.
__device__ __forceinline__ float wave_reduce_wmma(float p) {
    v2f a; a.x = p;    a.y = 0.0f;
    v2f b; b.x = 1.0f; b.y = 1.0f;
    v8f c = {};
    v8f d = __builtin_amdgcn_wmma_f32_16x16x4_f32(
        /*neg_a=*/false, a, /*neg_b=*/false, b,
        /*c_mod=*/(short)0, c, /*reuse_a=*/false, /*reuse_b=*/false);
    float s = d[0] + d[1] + d[2] + d[3] + d[4] + d[5] + d[6] + d[7];
    s += __shfl_xor(s, 16);
    return s;
}

__global__ void focal_main_kernel(const float* __restrict__ cls,   // [B, A, C]
                                  const float* __restrict__ reg,   // [B, A, 4]
                                  const float* __restrict__ anc,   // [A, 4] (y1,x1,y2,x2)
                                  const float* __restrict__ ann,   // [B, M, 5] (x1,y1,x2,y2,label)
                                  float* __restrict__ acc,         // [3*B]: cls | reg | npos
                                  int A, int B, int M, int C) {
    const int b = blockIdx.y;
    const int a = blockIdx.x * blockDim.x + threadIdx.x;
    const bool lane_valid = (a < A);

    __shared__ float s_ann[MAX_M * 5];
    const int tot = M * 5;

    // ---- stage annotations via async global->LDS DMA (gfx1250 ASYNCcnt path).
    // LDS aperture lives in the high 32 bits of a flat address, so addr[31:0]
    // of the generic pointer is the raw LDS byte offset the DMA needs.
    {
        const float* gsrc = ann + (size_t)b * tot;
        const unsigned lds_base = (unsigned)(uintptr_t)(void*)s_ann;
        for (int i = threadIdx.x; i < tot; i += blockDim.x) {
            const unsigned           lds_addr = lds_base + 4u * (unsigned)i;
            const unsigned long long gaddr    = (unsigned long long)(uintptr_t)(gsrc + i);
            asm volatile("global_load_async_to_lds_b32 %0, %1, off"
                         :: "v"(lds_addr), "v"(gaddr) : "memory");
        }
#if __has_builtin(__builtin_amdgcn_s_wait_asynccnt)
        __builtin_amdgcn_s_wait_asynccnt(0);
#else
        asm volatile("s_wait_asynccnt 0x0" ::: "memory");
#endif
    }
    __syncthreads();

    float cls_partial = 0.0f;
    float reg_partial = 0.0f;
    int   is_pos      = 0;

    if (lane_valid) {
        // Anchor geometry: anchor = [y1, x1, y2, x2]
        const float ay1 = anc[a * 4 + 0];
        const float ax1 = anc[a * 4 + 1];
        const float ay2 = anc[a * 4 + 2];
        const float ax2 = anc[a * 4 + 3];
        const float aw  = ax2 - ax1;
        const float ah  = ay2 - ay1;
        const float acx = ax1 + 0.5f * aw;
        const float acy = ay1 + 0.5f * ah;
        const float anchor_area = aw * ah;

        const float* crow = cls + ((size_t)b * A + a) * (size_t)C;
        __builtin_prefetch(crow, 0, 0);   // global_prefetch_b8: cover IoU-loop latency

        // IoU max / argmax over M annotations (annotation box = [x1,y1,x2,y2])
        float best = -2e9f;
        int   barg = 0;
        for (int j = 0; j < M; ++j) {
            const float bx1 = s_ann[j * 5 + 0];
            const float by1 = s_ann[j * 5 + 1];
            const float bx2 = s_ann[j * 5 + 2];
            const float by2 = s_ann[j * 5 + 3];
            const float lab = s_ann[j * 5 + 4];
            float iw = fminf(ax2, bx2) - fmaxf(ax1, bx1);
            float ih = fminf(ay2, by2) - fmaxf(ay1, by1);
            iw = fmaxf(iw, 0.0f);
            ih = fmaxf(ih, 0.0f);
            const float inter = iw * ih;
            const float area  = (bx2 - bx1) * (by2 - by1);
            const float ua    = fmaxf(anchor_area + area - inter, 1e-8f);
            const float iou   = (lab != -1.0f) ? (inter / ua) : -1e9f;
            if (iou > best) { best = iou; barg = j; }   // strict '>' => first-max, like argmax
        }

        const bool pos = (best >= 0.5f);
        const bool neg = (best <  0.4f);
        is_pos = pos ? 1 : 0;

        const int lab = (int)fmaxf(s_ann[barg * 5 + 4], 0.0f);

        // ---- classification focal loss (ignore band contributes nothing) ----
        if (pos || neg) {
            float csum = 0.0f;
            for (int k = 0; k < C; ++k) {
                float p = crow[k];
                p = fminf(fmaxf(p, 1e-4f), 1.0f - 1e-4f);
                const bool one = pos && (k == lab);
                const float af  = one ? 0.25f : 0.75f;
                const float q   = one ? (1.0f - p) : p;          // (gamma = 2 -> square)
                const float bce = one ? -__logf(p) : -__logf(1.0f - p);
                csum = fmaf(af * q * q, bce, csum);
            }
            cls_partial = csum;
        }

        // ---- regression smooth-L1 on positives ----
        if (pos) {
            const float bx1 = s_ann[barg * 5 + 0];
            const float by1 = s_ann[barg * 5 + 1];
            const float bx2 = s_ann[barg * 5 + 2];
            const float by2 = s_ann[barg * 5 + 3];
            float gw = bx2 - bx1;
            float gh = by2 - by1;
            const float gcx = bx1 + 0.5f * gw;   // centers from unclamped w/h (matches ref)
            const float gcy = by1 + 0.5f * gh;
            gw = fmaxf(gw, 1.0f);
            gh = fmaxf(gh, 1.0f);
            const float rt0 = (gcy - acy) / ah;
            const float rt1 = (gcx - acx) / aw;
            const float rt2 = __logf(gh / ah);
            const float rt3 = __logf(gw / aw);

            const float* rrow = reg + ((size_t)b * A + a) * 4;
            const float r0 = rrow[0], r1 = rrow[1], r2 = rrow[2], r3 = rrow[3];

            float s = 0.0f, dd;
            dd = fabsf(rt0 - r0); s += (dd <= (1.0f / 9.0f)) ? 4.5f * dd * dd : dd - (0.5f / 9.0f);
            dd = fabsf(rt1 - r1); s += (dd <= (1.0f / 9.0f)) ? 4.5f * dd * dd : dd - (0.5f / 9.0f);
            dd = fabsf(rt2 - r2); s += (dd <= (1.0f / 9.0f)) ? 4.5f * dd * dd : dd - (0.5f / 9.0f);
            dd = fabsf(rt3 - r3); s += (dd <= (1.0f / 9.0f)) ? 4.5f * dd * dd : dd - (0.5f / 9.0f);
            reg_partial = s;
        }
    }

    // ---- uniform control flow from here: EXEC all-ones as WMMA requires ----
    const float wc = wave_reduce_wmma(cls_partial);
    const float wr = wave_reduce_wmma(reg_partial);
    const unsigned long long bal = __ballot(is_pos);
    const int wn = __popcll(bal);

    if ((threadIdx.x & 31) == 0) {
        atomicAdd(&acc[b],          wc);
        atomicAdd(&acc[B + b],      wr);
        atomicAdd(&acc[2 * B + b],  (float)wn);
    }
}

__global__ void focal_finalize_kernel(const float* __restrict__ acc,
                                      float* __restrict__ out, int B) {
    if (threadIdx.x == 0 && blockIdx.x == 0) {
        float cls_m = 0.0f, reg_m = 0.0f;
        for (int b = 0; b < B; ++b) {
            const float np = fmaxf(acc[2 * B + b], 1.0f);
            cls_m += acc[b] / np;
            reg_m += acc[B + b] / (np * 4.0f);   // zero-sum when num_pos==0 (matches ref where())
        }
        out[0] = cls_m / (float)B;
        out[1] = (reg_m / (float)B) * 50.0f;
    }
}

extern "C" void kernel_launch(void* const* d_in, const int* in_sizes, int n_in,
                              void* d_out, int out_size, void* d_ws, size_t ws_size,
                              hipStream_t stream) {
    const float* cls = (const float*)d_in[0];   // [B, A, C]
    const float* reg = (const float*)d_in[1];   // [B, A, 4]
    const float* anc = (const float*)d_in[2];   // [1, A, 4]
    const float* ann = (const float*)d_in[3];   // [B, M, 5]
    float* out = (float*)d_out;

    const int A = in_sizes[2] / 4;
    const int B = in_sizes[1] / (A * 4);
    const int M = in_sizes[3] / (B * 5);
    const int C = (int)((long long)in_sizes[0] / ((long long)B * A));

    float* acc = (float*)d_ws;                  // 3*B floats of scratch
    hipMemsetAsync(acc, 0, (size_t)(3 * B) * sizeof(float), stream);

    const int threads = 256;                    // 8 waves per block (wave32)
    dim3 grid((A + threads - 1) / threads, B);
    focal_main_kernel<<<grid, threads, 0, stream>>>(cls, reg, anc, ann, acc, A, B, M, C);
    focal_finalize_kernel<<<1, 32, 0, stream>>>(acc, out, B);
}